// TaggerGreedyDecoder_64802466562822
// MI455X (gfx1250) — compile-verified
//
#include <hip/hip_runtime.h>
#include <cstdint>

#define B_ 256
#define T_ 512
#define N_ 128
#define GO_ 1
#define EOS_ 2
#define NEG_ -10000.0f

typedef __attribute__((address_space(1))) int as1_int;
typedef __attribute__((address_space(3))) int as3_int;

// ---------------------------------------------------------------------------
// gfx1250 async global->LDS copy (cp.async idiom). Tracked by ASYNCcnt.
// ---------------------------------------------------------------------------
__device__ __forceinline__ void async_g2l_b32(const float* g, float* l) {
#if __has_builtin(__builtin_amdgcn_global_load_async_to_lds_b32)
  __builtin_amdgcn_global_load_async_to_lds_b32(
      (as1_int*)(uintptr_t)g,
      (as3_int*)l,
      /*offset=*/0, /*cpol=*/0);
#else
  unsigned loff = (unsigned)(uintptr_t)(as3_int*)l;
  asm volatile("global_load_async_to_lds_b32 %0, %1, off"
               :: "v"(loff), "v"(g)
               : "memory");
#endif
}

template <int N>
__device__ __forceinline__ void wait_async_le() {
#if __has_builtin(__builtin_amdgcn_s_wait_asynccnt)
  __builtin_amdgcn_s_wait_asynccnt(N);
#else
  asm volatile("s_wait_asynccnt %0" :: "i"(N) : "memory");
#endif
}

// ---------------------------------------------------------------------------
// Kernel A: per-row log-sum-exp of unaries [B*T rows of N=128 floats].
// One wave32 per row; float4 per lane; shfl_xor tree reductions.
// A uniform shift never changes any argmax, so the serial Viterbi kernel only
// needs Sum(lse over t<len) as a scalar score correction.
// ---------------------------------------------------------------------------
__global__ __launch_bounds__(256) void lse_rows_kernel(
    const float* __restrict__ unaries, float* __restrict__ lse, int rows) {
  int row  = blockIdx.x * 8 + (threadIdx.x >> 5);
  int lane = threadIdx.x & 31;
  if (row >= rows) return;
  float4 v = ((const float4*)(unaries + (size_t)row * N_))[lane];
  float m = fmaxf(fmaxf(v.x, v.y), fmaxf(v.z, v.w));
#pragma unroll
  for (int off = 16; off > 0; off >>= 1) m = fmaxf(m, __shfl_xor(m, off));
  float s = __expf(v.x - m) + __expf(v.y - m) + __expf(v.z - m) + __expf(v.w - m);
#pragma unroll
  for (int off = 16; off > 0; off >>= 1) s += __shfl_xor(s, off);
  if (lane == 0) lse[row] = m + __logf(s);
}

// ---------------------------------------------------------------------------
// Kernel B: one workgroup per batch element. thread = current state.
//  - trans row (128 f32) fully unrolled into VGPRs
//  - alphas double-buffered in LDS -> exactly ONE barrier per serial step
//  - alpha broadcast via ds_load_b128 (same-address, conflict-free)
//  - backpointers live entirely in LDS (64KB) -> LDS-latency backtrace
//  - unary rows streamed through a 4-deep async global->LDS ring,
//    3 rows in flight, s_wait_asynccnt<=2 (no barrier needed: each lane
//    reads only the element its own async wrote)
// ---------------------------------------------------------------------------
__global__ __launch_bounds__(128) void viterbi_kernel(
    const float* __restrict__ unaries,   // [B, T, N]
    const float* __restrict__ trans,     // [N(cur), N(prev)]
    const int*   __restrict__ lengths,   // [B]
    const float* __restrict__ lse,       // [B, T]
    float* __restrict__ out_path,        // [B, T] (float-encoded tags)
    float* __restrict__ out_score)       // [B]
{
  __shared__ unsigned char s_bp[T_ * N_];   // 64 KB backpointers, whole lattice
  __shared__ float s_alpha[2][N_];          // double buffer
  __shared__ float s_u[4][N_];              // async ring buffer
  __shared__ float s_red[4];
  __shared__ int   s_redi[4];
  __shared__ float s_part[N_];

  const int tid = threadIdx.x;
  const int b   = blockIdx.x;
  int len = lengths[b];
  if (len < 1)  len = 1;
  if (len > T_) len = T_;

  // trans row for this 'cur' state -> registers (fully unrolled, const idx)
  float tr[N_];
  {
    const float4* trow = (const float4*)(trans + (size_t)tid * N_);
#pragma unroll
    for (int i = 0; i < N_ / 4; ++i) {
      float4 v = trow[i];
      tr[4*i+0] = v.x; tr[4*i+1] = v.y; tr[4*i+2] = v.z; tr[4*i+3] = v.w;
    }
  }

  // init alphas: log_softmax of (0 at GO, NEG elsewhere) == same values (lse~0)
  s_alpha[0][tid] = (tid == GO_) ? 0.0f : NEG_;

  const float* ub = unaries + (size_t)b * T_ * N_;
  // prologue: put rows 0..2 in flight
#pragma unroll
  for (int r = 0; r < 3; ++r)
    if (r < len) async_g2l_b32(ub + (size_t)r * N_ + tid, &s_u[r][tid]);

  for (int t = 0; t < len; ++t) {
    wait_async_le<2>();                     // oldest in-flight row (t) landed
    float u = s_u[t & 3][tid];              // own element; own async -> no barrier
    if (t + 3 < len)
      async_g2l_b32(ub + (size_t)(t + 3) * N_ + tid, &s_u[(t + 3) & 3][tid]);

    __syncthreads();                        // publishes alphas written at t-1
                                            // (and the t=0 init)

    // max-plus MV product over prev, first-index tie-break (matches argmax)
    const float* al = s_alpha[t & 1];
    float m = -3.0e38f;
    int bp = 0;
#pragma unroll
    for (int p = 0; p < N_; p += 4) {
      float4 a = *(const float4*)(al + p);   // ds_load_b128 broadcast
      float v0 = a.x + tr[p + 0];
      float v1 = a.y + tr[p + 1];
      float v2 = a.z + tr[p + 2];
      float v3 = a.w + tr[p + 3];
      if (v0 > m) { m = v0; bp = p;     }
      if (v1 > m) { m = v1; bp = p + 1; }
      if (v2 > m) { m = v2; bp = p + 2; }
      if (v3 > m) { m = v3; bp = p + 3; }
    }
    s_bp[t * N_ + tid] = (unsigned char)bp;
    s_alpha[(t + 1) & 1][tid] = m + u;      // other buffer: no barrier needed;
                                            // next iter's barrier publishes it
  }
  __syncthreads();

  // terminal = alpha + trans0[EOS][:]   (own element -> no extra barrier)
  float term = s_alpha[len & 1][tid] + trans[(size_t)EOS_ * N_ + tid];
  float bm = term; int bi = tid;
#pragma unroll
  for (int off = 16; off > 0; off >>= 1) {  // wave32 argmax tree
    float om = __shfl_down(bm, off);
    int   oi = __shfl_down(bi, off);
    if (om > bm || (om == bm && oi < bi)) { bm = om; bi = oi; }
  }
  if ((tid & 31) == 0) { s_red[tid >> 5] = bm; s_redi[tid >> 5] = bi; }

  // deterministic partial sums of lse over t < len
  const float* lb = lse + (size_t)b * T_;
  float part = 0.0f;
  for (int t = tid; t < len; t += N_) part += lb[t];
  s_part[tid] = part;
  __syncthreads();

  if (tid == 0) {
    float fm = s_red[0]; int fi = s_redi[0];
#pragma unroll
    for (int w = 1; w < 4; ++w) {
      float om = s_red[w]; int oi = s_redi[w];
      if (om > fm || (om == fm && oi < fi)) { fm = om; fi = oi; }
    }
    float lsum = 0.0f;
    for (int i = 0; i < N_; ++i) lsum += s_part[i];
    out_score[b] = fm - lsum;

    // backtrace entirely from LDS
    float* op = out_path + (size_t)b * T_;
    int bt = fi;
    op[len - 1] = (float)bt;
    for (int tp = len - 1; tp >= 1; --tp) {
      bt = s_bp[tp * N_ + bt];
      op[tp - 1] = (float)bt;
    }
  }

  // masked tail -> 0 (positions t >= len), disjoint from backtrace region
  float* op = out_path + (size_t)b * T_;
  for (int t = len + tid; t < T_; t += N_) op[t] = 0.0f;
}

extern "C" void kernel_launch(void* const* d_in, const int* in_sizes, int n_in,
                              void* d_out, int out_size, void* d_ws, size_t ws_size,
                              hipStream_t stream) {
  (void)in_sizes; (void)n_in; (void)out_size; (void)ws_size;
  const float* unaries = (const float*)d_in[0];
  const float* trans   = (const float*)d_in[1];
  const int*   lengths = (const int*)d_in[2];
  float* out_path  = (float*)d_out;                 // [B,T] tags as float
  float* out_score = out_path + (size_t)B_ * T_;    // [B]
  float* lse       = (float*)d_ws;                  // [B*T] f32 = 512 KB

  const int rows = B_ * T_;
  lse_rows_kernel<<<rows / 8, 256, 0, stream>>>(unaries, lse, rows);
  viterbi_kernel<<<B_, 128, 0, stream>>>(unaries, trans, lengths, lse,
                                         out_path, out_score);
}